// ResGNN_42606075576950
// MI455X (gfx1250) — compile-verified
//
#include <hip/hip_runtime.h>

// ResGNN on MI455X (gfx1250): memory-bound streaming of the 256MB env tensor.
// 3 full env passes (layer1, layer2, policy) + tiny glue kernels.
// Matmuls use V_WMMA_F32_16X16X4_F32 (wave32, A=16x4 f32 in 2 VGPRs).

#define NB 32
#define NN 1024

typedef __attribute__((ext_vector_type(2))) float v2f;
typedef __attribute__((ext_vector_type(4))) float v4f;
typedef __attribute__((ext_vector_type(8))) float v8f;

// ---------------------------------------------------------------------------
// Streaming WGCN pass:  Xout[b,i,f] = relu( sum_j (A*W)[b,i,j] * GT[b,f,j] + bias[f] )
// GT is stored (B, NT*16, N)  (f-major => K-contiguous for the B operand).
// One wave = one 16-row strip. NT f-tiles of 16 columns each.
// A tile layout (ISA 7.12.2, 32-bit A 16x4): lanes 0-15 rows, VGPR0=K0/K2,
// VGPR1=K1/K3 split by lane half -> lane half loads K {0,1} or {2,3}.
// ---------------------------------------------------------------------------
template <int NT, int FVALID, int FSTRIDE>
__global__ __launch_bounds__(128) void wgcn_pass(
    const float* __restrict__ env,   // (B,N,N,2)
    const float* __restrict__ GT,    // (B,NT*16,N)
    const float* __restrict__ bias,  // (FVALID)
    float* __restrict__ Xout)        // (B,N,FSTRIDE)
{
    const int lane  = threadIdx.x & 31;
    const int wave  = threadIdx.x >> 5;
    const int b     = blockIdx.x >> 4;                 // 16 blocks per batch
    const int strip = ((blockIdx.x & 15) << 2) + wave; // 0..63
    const int row0  = strip << 4;
    const int hf    = lane >> 4;   // lane half: 0 -> K{0,1}, 1 -> K{2,3}
    const int r     = lane & 15;   // row within strip (A), col within tile (B)

    const float* erow = env + (size_t)(b * NN + row0 + r) * (2 * NN);
    v8f acc[NT] = {};

    for (int j = 0; j < NN; j += 4) {
        // 4 floats = (A,W) pairs at j0 and j0+1 for this lane's K slots.
        v4f p = __builtin_nontemporal_load(
            (const v4f*)(erow + 2 * (j + 2 * hf)));
        v2f a;
        a.x = p.x * p.y;   // M[row, j0]
        a.y = p.z * p.w;   // M[row, j0+1]
#pragma unroll
        for (int t = 0; t < NT; ++t) {
            // B tile: col = lane&15, K rows contiguous in GT (hits L2).
            const float* gp =
                GT + (size_t)(b * (NT * 16) + t * 16 + r) * NN + j + 2 * hf;
            v2f g = *(const v2f*)gp;
            acc[t] = __builtin_amdgcn_wmma_f32_16x16x4_f32(
                false, a, false, g, (short)0, acc[t], false, false);
        }
    }

#pragma unroll
    for (int t = 0; t < NT; ++t) {
        const int f  = t * 16 + r;
        const float bf = (f < FVALID) ? bias[f] : 0.f;
#pragma unroll
        for (int v = 0; v < 8; ++v) {
            const int orow = row0 + v + 8 * hf;  // C/D layout: VGPR v -> row v / v+8
            Xout[(size_t)(b * NN + orow) * FSTRIDE + f] =
                fmaxf(acc[t][v] + bf, 0.f);
        }
    }
}

// ---------------------------------------------------------------------------
// Layer 0 exploits one-hot X: only column state[b] of A*W is needed.
// Produces G1T[b,f,j] = sum_c relu(M[b,j]*k0[c]+b0[c]) * k1[c,f], padded to 32.
// ---------------------------------------------------------------------------
__global__ __launch_bounds__(256) void k_init_g1(
    const float* __restrict__ env, const int* __restrict__ state,
    const float* __restrict__ k0, const float* __restrict__ b0,
    const float* __restrict__ k1, float* __restrict__ G1T)
{
    const int gid = blockIdx.x * 256 + threadIdx.x;  // b*NN + j
    const int b = gid >> 10, j = gid & (NN - 1);
    const int s = state[b];
    const float* p = env + ((size_t)(b * NN + j) * NN + s) * 2;
    const float m = p[0] * p[1];
    float x1[16];
#pragma unroll
    for (int c = 0; c < 16; ++c) x1[c] = fmaxf(m * k0[c] + b0[c], 0.f);
    for (int f = 0; f < 24; ++f) {
        float g = 0.f;
#pragma unroll
        for (int c = 0; c < 16; ++c) g += x1[c] * k1[c * 24 + f];
        G1T[(size_t)(b * 32 + f) * NN + j] = g;
    }
    for (int f = 24; f < 32; ++f) G1T[(size_t)(b * 32 + f) * NN + j] = 0.f;
}

// G2T[b,f,j] = sum_{c<24} X2[b,j,c] * k2[c,f]   (f<64)
__global__ __launch_bounds__(256) void k_build_g2(
    const float* __restrict__ X2, const float* __restrict__ k2,
    float* __restrict__ G2T)
{
    const int gid = blockIdx.x * 256 + threadIdx.x;  // b*NN + j
    const int b = gid >> 10, j = gid & (NN - 1);
    float x[24];
    const float* xr = X2 + (size_t)gid * 32;
#pragma unroll
    for (int c = 0; c < 24; ++c) x[c] = xr[c];
    for (int f = 0; f < 64; ++f) {
        float g = 0.f;
#pragma unroll
        for (int c = 0; c < 24; ++c) g += x[c] * k2[c * 64 + f];
        G2T[(size_t)(b * 64 + f) * NN + j] = g;
    }
}

// y[b,j] = X3[b,j,:] . pol_k ; also capture proj row X3[b,state[b],:]
__global__ __launch_bounds__(256) void k_build_y(
    const float* __restrict__ X3, const int* __restrict__ state,
    const float* __restrict__ polk, float* __restrict__ y,
    float* __restrict__ proj)
{
    const int gid = blockIdx.x * 256 + threadIdx.x;
    const int b = gid >> 10, j = gid & (NN - 1);
    const float* xr = X3 + (size_t)gid * 64;
    float acc = 0.f;
#pragma unroll
    for (int c = 0; c < 64; ++c) acc += xr[c] * polk[c];
    y[gid] = acc;
    if (j == state[b]) {
        for (int c = 0; c < 64; ++c) proj[b * 64 + c] = xr[c];
    }
}

// Policy pass: Z[b,i] = relu( sum_j M[b,i,j]*y[b,j] + pol_b ). F=1 -> VALU dot.
__global__ __launch_bounds__(256) void k_pol(
    const float* __restrict__ env, const float* __restrict__ y,
    const float* __restrict__ pol_b, float* __restrict__ Z)
{
    const int lane = threadIdx.x & 31;
    const int wv   = threadIdx.x >> 5;
    const int rowg = blockIdx.x * 8 + wv;  // b*NN + i
    const int b    = rowg >> 10;
    const float* er = env + (size_t)rowg * (2 * NN);
    const float* yb = y + b * NN;
    float s = 0.f;
    for (int j = lane; j < NN; j += 32) {
        v2f p = __builtin_nontemporal_load((const v2f*)(er + 2 * j));
        s += p.x * p.y * yb[j];
    }
#pragma unroll
    for (int off = 16; off > 0; off >>= 1) s += __shfl_xor(s, off, 32);
    if (lane == 0) Z[rowg] = fmaxf(s + pol_b[0], 0.f);
}

// Value head + softmax; one block per batch.
__global__ __launch_bounds__(1024) void k_head(
    const float* __restrict__ proj, const float* __restrict__ Z,
    const float* __restrict__ gamma, const float* __restrict__ beta,
    const float* __restrict__ mean, const float* __restrict__ var,
    const float* __restrict__ fcw, const float* __restrict__ fcb,
    const float* __restrict__ valw, const float* __restrict__ valb,
    float* __restrict__ out)  // [0..31]=v, [32 + b*N + i]=pi
{
    __shared__ float red[1024];
    __shared__ float hbn[64];
    __shared__ float h[128];
    const int b = blockIdx.x, t = threadIdx.x;

    const float z = Z[b * NN + t];
    red[t] = z; __syncthreads();
    for (int off = 512; off > 0; off >>= 1) {
        if (t < off) red[t] = fmaxf(red[t], red[t + off]);
        __syncthreads();
    }
    const float m = red[0]; __syncthreads();
    const float e = expf(z - m);
    red[t] = e; __syncthreads();
    for (int off = 512; off > 0; off >>= 1) {
        if (t < off) red[t] += red[t + off];
        __syncthreads();
    }
    out[32 + b * NN + t] = e * (1.f / red[0]);

    if (t < 64)
        hbn[t] = (proj[b * 64 + t] - mean[t]) * rsqrtf(var[t] + 1e-3f) * gamma[t]
                 + beta[t];
    __syncthreads();
    if (t < 128) {
        float a = fcb[t];
        for (int f = 0; f < 64; ++f) a += hbn[f] * fcw[f * 128 + t];
        h[t] = fmaxf(a, 0.f);
    }
    __syncthreads();
    if (t == 0) {
        float a = valb[0];
        for (int f = 0; f < 128; ++f) a += h[f] * valw[f];
        out[b] = tanhf(a);
    }
}

extern "C" void kernel_launch(void* const* d_in, const int* in_sizes, int n_in,
                              void* d_out, int out_size, void* d_ws, size_t ws_size,
                              hipStream_t stream)
{
    (void)in_sizes; (void)n_in; (void)out_size; (void)ws_size;
    const float* env   = (const float*)d_in[0];
    const int*   state = (const int*)  d_in[1];
    const float* k0    = (const float*)d_in[2];
    const float* b0    = (const float*)d_in[3];
    const float* k1    = (const float*)d_in[4];
    const float* b1    = (const float*)d_in[5];
    const float* k2    = (const float*)d_in[6];
    const float* b2    = (const float*)d_in[7];
    const float* gamma = (const float*)d_in[8];
    const float* beta  = (const float*)d_in[9];
    const float* mean  = (const float*)d_in[10];
    const float* var   = (const float*)d_in[11];
    const float* fcw   = (const float*)d_in[12];
    const float* fcb   = (const float*)d_in[13];
    const float* valw  = (const float*)d_in[14];
    const float* valb  = (const float*)d_in[15];
    const float* polk  = (const float*)d_in[16];
    const float* polb  = (const float*)d_in[17];
    float* out = (float*)d_out;

    // Workspace layout (floats), total ~25.4 MB.
    float* ws  = (float*)d_ws;
    float* G1T = ws;                             // 32*32*1024
    float* X2  = G1T + (size_t)NB * 32 * NN;     // 32*1024*32
    float* G2T = X2  + (size_t)NB * NN * 32;     // 32*64*1024
    float* X3  = G2T + (size_t)NB * 64 * NN;     // 32*1024*64
    float* Y   = X3  + (size_t)NB * NN * 64;     // 32*1024
    float* PRJ = Y   + (size_t)NB * NN;          // 32*64
    float* Zb  = PRJ + (size_t)NB * 64;          // 32*1024

    k_init_g1 <<<(NB * NN) / 256, 256, 0, stream>>>(env, state, k0, b0, k1, G1T);
    wgcn_pass<2, 24, 32><<<NB * 16, 128, 0, stream>>>(env, G1T, b1, X2);
    k_build_g2<<<(NB * NN) / 256, 256, 0, stream>>>(X2, k2, G2T);
    wgcn_pass<4, 64, 64><<<NB * 16, 128, 0, stream>>>(env, G2T, b2, X3);
    k_build_y <<<(NB * NN) / 256, 256, 0, stream>>>(X3, state, polk, Y, PRJ);
    k_pol     <<<(NB * NN) / 8,   256, 0, stream>>>(env, Y, polb, Zb);
    k_head    <<<NB, 1024, 0, stream>>>(PRJ, Zb, gamma, beta, mean, var,
                                        fcw, fcb, valw, valb, out);
}